// Conv2D_29703993819153
// MI455X (gfx1250) — compile-verified
//
#include <hip/hip_runtime.h>

typedef __attribute__((ext_vector_type(2))) float v2f;
typedef __attribute__((ext_vector_type(8))) float v8f;
typedef __attribute__((ext_vector_type(4))) unsigned int v4u;
typedef __attribute__((ext_vector_type(8))) int v8i;
typedef __attribute__((ext_vector_type(4))) int v4i;

#define H 4096
#define W 4096
#define KH 15
#define KW 15
#define OH (H - KH + 1) /* 4082 */
#define OW (W - KW + 1) /* 4082 */

#define WAVES_PER_BLOCK 8
#define PATCH_ROWS 30   /* 16 + 15 - 1 */
#define PATCH_STRIDE 46 /* TDM writes rows contiguously; 46*4B is 8B-aligned */
#define PATCH_SLACK 8   /* kc=11 chunk may read 2 floats past the patch end */
#define WROW 48         /* padded weight row covers shifts s in [-16, 31] */

#define USE_TDM (__has_builtin(__builtin_amdgcn_tensor_load_to_lds) && \
                 __has_builtin(__builtin_amdgcn_s_wait_tensorcnt))

__global__ __launch_bounds__(256) void conv15_wmma_f32(
    const float* __restrict__ x, const float* __restrict__ wk,
    const float* __restrict__ bias, float* __restrict__ out) {
  __shared__ float s_patch[WAVES_PER_BLOCK][PATCH_ROWS * PATCH_STRIDE + PATCH_SLACK];
  __shared__ float s_w[KH * WROW];

  const int tid = threadIdx.x;
  const int lane = tid & 31;
  const int wid = __builtin_amdgcn_readfirstlane(tid >> 5);  // wave-uniform

  // One wave -> one 16x32 output tile.
  const int task = blockIdx.x * WAVES_PER_BLOCK + wid;  // 0..32767
  const int tp = task >> 7;                             // 0..255
  const int tq = task & 127;                            // 0..127
  const int p0 = tp * 16;
  const int q0 = tq * 32;

  float* patch = s_patch[wid];

#if USE_TDM
  // ---- Tensor Data Mover: async DMA of the 30x46 f32 tile into LDS.
  // D# group 0: count=1 | lds_addr | global_addr[56:0] | type=2.
  {
    // Generic LDS pointer: aperture in addr[63:32], byte offset in addr[31:0].
    unsigned int lds_off = (unsigned int)(unsigned long long)(void*)patch;
    unsigned long long gaddr =
        (unsigned long long)(const void*)x +
        4ull * ((unsigned long long)p0 * W + (unsigned long long)q0);

    v4u g0;
    g0[0] = 1u;                                        // count=1, user mode
    g0[1] = lds_off;                                   // lds_addr (bytes)
    g0[2] = (unsigned int)gaddr;                       // global_addr[31:0]
    g0[3] = (unsigned int)(gaddr >> 32) | (2u << 30);  // addr[56:32] | type=2

    // D# group 1 (bit-packed per ISA 8.4):
    v8i g1;
    g1[0] = 0x00020000;           // wg_mask=0, data_size=2 (4B)
    g1[1] = (W & 0xFFFF) << 16;   // tensor_dim0[15:0] at bits[63:48]
    g1[2] = ((W >> 16) & 0xFFFF) | ((H & 0xFFFF) << 16);  // dim0 hi | dim1 lo
    g1[3] = ((H >> 16) & 0xFFFF) | (PATCH_STRIDE << 16);  // dim1 hi | tile_dim0
    g1[4] = PATCH_ROWS;           // tile_dim1=30, tile_dim2=0
    g1[5] = W;                    // tensor_dim0_stride[31:0]
    g1[6] = 0;                    // dim0_stride hi | dim1_stride lo
    g1[7] = 0;                    // dim1_stride hi
    v4i gz4 = {0, 0, 0, 0};       // groups 2/3 unused (2D tensor)
    v8i gz8 = {0, 0, 0, 0, 0, 0, 0, 0};
    __builtin_amdgcn_tensor_load_to_lds(g0, g1, gz4, gz4, gz8, 0);
  }
#else
  // Fallback: scalar staging with edge clamp (clamped cells only feed
  // outputs that get masked on store).
  for (int t = lane; t < PATCH_ROWS * PATCH_STRIDE; t += 32) {
    int row = t / PATCH_STRIDE;
    int col = t % PATCH_STRIDE;
    int gr = min(p0 + row, H - 1);
    int gc = min(q0 + col, W - 1);
    patch[t] = x[gr * W + gc];
  }
#endif

  // Stage zero-padded kernel rows (overlaps with the TDM transfer):
  // s_w[r*48 + 16 + s] = W[r,s], 0 elsewhere.
  for (int t = tid; t < KH * WROW; t += 256) {
    int r = t / WROW;
    int c = t % WROW;
    int s = c - 16;
    s_w[t] = (s >= 0 && s < KW) ? wk[r * KW + s] : 0.0f;
  }
  __syncthreads();
#if USE_TDM
  __builtin_amdgcn_s_wait_tensorcnt(0);
  asm volatile("" ::: "memory");  // keep LDS reads below the tensor wait
#endif

  const int j = lane & 15;            // A: M row; B: N column
  const int klane = (lane >> 4) * 2;  // K sub-offset owned by this half-wave

  v8f acc0 = {};
  v8f acc1 = {};

#pragma unroll 1
  for (int r = 0; r < KH; ++r) {
    // 8 B-fragments for this kernel row: B[k, j] = W[r, k - j] (zero-padded).
    const float* wrow = &s_w[r * WROW + 16 + klane - j];
    v2f bf[8];
#pragma unroll
    for (int kc = 0; kc < 8; ++kc) {
      bf[kc].x = wrow[4 * kc + 0];
      bf[kc].y = wrow[4 * kc + 1];
    }
    // A slab: row (i + r) of the patch; acc1 reuses bf with a +16-col A shift.
    const float* arow = &patch[(j + r) * PATCH_STRIDE + klane];
#pragma unroll
    for (int kc = 0; kc < 12; ++kc) {
      v2f a;
      a.x = arow[4 * kc + 0];
      a.y = arow[4 * kc + 1];
      if (kc < 8)
        acc0 = __builtin_amdgcn_wmma_f32_16x16x4_f32(
            false, a, false, bf[kc], (short)0, acc0, false, false);
      if (kc >= 4)
        acc1 = __builtin_amdgcn_wmma_f32_16x16x4_f32(
            false, a, false, bf[kc - 4], (short)0, acc1, false, false);
    }
  }

  // Store: D layout — lane half picks M base (0 or 8), VGPR v adds M, N = j.
  const float b0 = bias[0];
  const int hi = lane >> 4;
#pragma unroll
  for (int v = 0; v < 8; ++v) {
    int p = p0 + 8 * hi + v;
    if (p < OH) {
      int q = q0 + j;  // q0 <= 4064, j <= 15 -> q <= 4079 < OW always
      out[p * OW + q] = acc0[v] + b0;
      int q2 = q + 16;
      if (q2 < OW) out[p * OW + q2] = acc1[v] + b0;
    }
  }
}

extern "C" void kernel_launch(void* const* d_in, const int* in_sizes, int n_in,
                              void* d_out, int out_size, void* d_ws,
                              size_t ws_size, hipStream_t stream) {
  const float* x = (const float*)d_in[0];     // 4096*4096 fp32
  const float* wk = (const float*)d_in[1];    // 15*15 fp32
  const float* bias = (const float*)d_in[2];  // 1 fp32
  float* out = (float*)d_out;                 // 4082*4082 fp32

  dim3 grid((256 * 128) / WAVES_PER_BLOCK);  // 4096 blocks
  dim3 block(256);                           // 8 waves (wave32)
  conv15_wmma_f32<<<grid, block, 0, stream>>>(x, wk, bias, out);
}